// HybridONNXPyTorchModel_21680994910678
// MI455X (gfx1250) — compile-verified
//
#include <hip/hip_runtime.h>
#include <math.h>
#include <stdint.h>

typedef float v2f __attribute__((ext_vector_type(2)));
typedef float v8f __attribute__((ext_vector_type(8)));

#define BATCH 16
#define TFRM  1000
#define DIN   80
#define CCEP  222
#define CPAD  224               // padded ccep row (pad cols are exact zeros)
#define HOPSZ 240
#define NSAMP (TFRM*HOPSZ)      // 240000
#define FFTN  1024
#define FSR   24000.0f
#define NHARM 200.0f
#define CONVLEN (HOPSZ+FFTN-1)  // 1263
#define NTILES 14               // ceil(222/16)
#define MGROUPS 125             // 16000 rows / (8 waves * 16)
#define XPAD  82                // LDS row pitch for x tile (odd in 8B units)

#if defined(__has_builtin)
#  if __has_builtin(__builtin_amdgcn_global_load_async_to_lds_b32)
#    define HAVE_ASYNC_LDS 1
#  endif
#endif
#ifndef HAVE_ASYNC_LDS
#  define HAVE_ASYNC_LDS 0
#endif

// ---------------------------------------------------------------------------
// Kernel 1: ccep = x @ W + b for both branches, f32 WMMA 16x16x4.
// Block = 8 waves sharing one ntile/branch; A tile (128x80) and B strip
// (80x16, zero-padded) staged in LDS -> inner loop is pure ds_load + wmma,
// no exec-mask churn.  Output stored padded to 224 cols (pads == 0.0).
// ---------------------------------------------------------------------------
__global__ __launch_bounds__(256)
void gemm_ccep_wmma(const float* __restrict__ x,
                    const float* __restrict__ Wh, const float* __restrict__ bh,
                    const float* __restrict__ Wn, const float* __restrict__ bn,
                    float* __restrict__ ccepH, float* __restrict__ ccepN)
{
    __shared__ float xs[128 * XPAD];   // 128 rows x 80 (pitch 82) = 41 KB
    __shared__ float wl[DIN * 16];     // 80 x 16 B strip = 5 KB

    const int tid  = threadIdx.x;
    const int wave = tid >> 5;
    const int lane = tid & 31;

    int bid = blockIdx.x;
    const int mgroup = bid % MGROUPS;  bid /= MGROUPS;
    const int ntile  = bid % NTILES;   bid /= NTILES;
    const int branch = bid;            // 0 or 1

    const float* W    = branch ? Wn : Wh;
    const float* bias = branch ? bn : bh;
    float*       outp = branch ? ccepN : ccepH;

    const int row0 = mgroup * 128;

    // stage A tile: coalesced float4 (global_load_b128) loads, 10 per thread
    for (int i = tid; i < 128 * (DIN / 4); i += 256) {
        const int r  = i / (DIN / 4);
        const int c4 = i - r * (DIN / 4);
        const float4 v = ((const float4*)(x + (size_t)(row0 + r) * DIN))[c4];
        float* d = &xs[r * XPAD + c4 * 4];
        d[0] = v.x; d[1] = v.y; d[2] = v.z; d[3] = v.w;
    }
    // stage B strip: 80x16, zero-padded past column 221 (5 elems/thread)
    for (int i = tid; i < DIN * 16; i += 256) {
        const int r = i >> 4, c = i & 15;
        const int col = ntile * 16 + c;
        wl[i] = (col < CCEP) ? W[(size_t)r * CCEP + col] : 0.0f;
    }
    __syncthreads();

    // A 16x4 f32 layout: lanes 0-15 K={0,1}, lanes 16-31 K={2,3}
    const int hi   = lane >> 4;
    const int l    = lane & 15;
    const int lrow = wave * 16 + l;

    v8f acc = {};
#pragma unroll
    for (int k0 = 0; k0 < DIN; k0 += 4) {
        const int ka = k0 + 2 * hi;
        v2f a, bm;
        a.x  = xs[lrow * XPAD + ka];
        a.y  = xs[lrow * XPAD + ka + 1];
        bm.x = wl[ka * 16 + l];
        bm.y = wl[(ka + 1) * 16 + l];
        acc = __builtin_amdgcn_wmma_f32_16x16x4_f32(
            false, a, false, bm, (short)0, acc, false, false);
    }

    // bias: clamp+mask instead of predicated load (no exec churn)
    const int  col  = ntile * 16 + l;
    const bool cok  = (col < CCEP);
    const float bv  = bias[cok ? col : (CCEP - 1)] * (cok ? 1.0f : 0.0f);

    // D layout: VGPR r -> M = r + 8*hi, N = lane&15.  Unconditional stores.
    const int mbase = row0 + wave * 16 + 8 * hi;
#pragma unroll
    for (int r = 0; r < 8; ++r)
        outp[(size_t)(mbase + r) * CPAD + col] = acc[r] + bv;
}

// ---------------------------------------------------------------------------
// Kernel 2: per-batch exclusive prefix sum over frames (the sample-level
// cumsum collapses analytically because f0 is piecewise-constant per hop).
// ---------------------------------------------------------------------------
__global__ void f0_prefix(const float* __restrict__ cf0, float* __restrict__ pre)
{
    const int b = threadIdx.x;
    if (b >= BATCH) return;
    float acc = 0.0f;
    for (int t = 0; t < TFRM; ++t) {
        pre[b * TFRM + t] = acc;
        acc += cf0[b * TFRM + t];
    }
}

// ---------------------------------------------------------------------------
// Kernel 3: band-limited impulse train (Dirichlet kernel) per sample.
// ---------------------------------------------------------------------------
__global__ __launch_bounds__(256)
void impulse_train(const float* __restrict__ cf0, const float* __restrict__ pre,
                   float* __restrict__ src)
{
    const long long i = (long long)blockIdx.x * 256 + threadIdx.x;
    if (i >= (long long)BATCH * NSAMP) return;
    const int b = (int)(i / NSAMP);
    const int n = (int)(i - (long long)b * NSAMP);
    const int t = n / HOPSZ;
    const int j = n - t * HOPSZ;

    const float  f0  = cf0[b * TFRM + t];
    const double cum = 240.0 * (double)pre[b * TFRM + t] + (double)(j + 1) * (double)f0;
    const float  ph  = (float)(6.283185307179586 * fmod(cum, 24000.0) / 24000.0);

    const float fmx  = fmaxf(f0, 1e-3f);
    const float K    = fminf(fmaxf(floorf(FSR / (2.0f * fmx)), 0.0f), NHARM);
    const float half = 0.5f * ph;
    const float s    = __sinf(half);
    float dir;
    if (fabsf(s) < 1e-4f) dir = K;
    else                  dir = __sinf((2.0f * K + 1.0f) * half) / (2.0f * s) - 0.5f;
    const float amp = sqrtf(2.0f * fmaxf(f0, 0.0f) / FSR);
    src[i] = (f0 > 0.0f) ? dir * amp : 0.0f;
}

// ---------------------------------------------------------------------------
// In-LDS 1024-point radix-2 complex FFT (bit-reverse + 10 stages).
// ---------------------------------------------------------------------------
__device__ __forceinline__ void fft1024_lds(float* re, float* im, int tid)
{
    for (int i = tid; i < FFTN; i += 256) {
        const int j = (int)(__brev((unsigned)i) >> 22);
        if (j > i) {
            float tr = re[i]; re[i] = re[j]; re[j] = tr;
            float ti = im[i]; im[i] = im[j]; im[j] = ti;
        }
    }
    __syncthreads();
    for (int len = 2; len <= FFTN; len <<= 1) {
        const int half = len >> 1;
        const float w0 = -6.28318530717958647f / (float)len;
        for (int bb = tid; bb < FFTN / 2; bb += 256) {
            const int grp = bb / half;
            const int pos = bb - grp * half;
            const int i0  = grp * len + pos;
            const int i1  = i0 + half;
            float si, ci;
            __sincosf(w0 * (float)pos, &si, &ci);
            const float xr = re[i1], xi = im[i1];
            const float tr = ci * xr - si * xi;
            const float ti = ci * xi + si * xr;
            const float ur = re[i0], ui = im[i0];
            re[i1] = ur - tr; im[i1] = ui - ti;
            re[i0] = ur + tr; im[i0] = ui + ti;
        }
        __syncthreads();
    }
}

// ---------------------------------------------------------------------------
// Kernel 4: per (frame, branch): FFT(ccep) -> exp -> IFFT (conj trick) = h,
// then direct time-varying FIR (240x1024) + overlap-add via f32 atomics.
// Source frame pulled in with an ASYNCcnt-tracked async LDS load when the
// toolchain exposes the gfx1250 builtin.
// ---------------------------------------------------------------------------
__global__ __launch_bounds__(256)
void ltv_filter(const float* __restrict__ ccepH, const float* __restrict__ ccepN,
                const float* __restrict__ harmSrc, const float* __restrict__ z,
                float* __restrict__ out)
{
    __shared__ float re[FFTN];
    __shared__ float im[FFTN];
    __shared__ float fr[HOPSZ];

    const int blk    = blockIdx.x;
    const int branch = (blk >= BATCH * TFRM) ? 1 : 0;
    const int ft     = blk - branch * BATCH * TFRM;   // 0..15999
    const int b      = ft / TFRM;
    const int t      = ft - b * TFRM;
    const int tid    = threadIdx.x;

    const float* ccep = branch ? ccepN : ccepH;
    const float* srcp = branch ? z : harmSrc;

    // source frame: async global->LDS copy if available
    if (tid < HOPSZ) {
        const float* g = srcp + (size_t)b * NSAMP + (size_t)t * HOPSZ + tid;
#if HAVE_ASYNC_LDS
        __builtin_amdgcn_global_load_async_to_lds_b32(
            (int*)g, (int*)&fr[tid], 0, 0);
#else
        fr[tid] = *g;
#endif
    }
    // cepstrum row (padded layout: cols 222/223 are exact zeros)
    for (int i = tid; i < FFTN; i += 256) {
        re[i] = (i < CPAD) ? ccep[(size_t)ft * CPAD + i] : 0.0f;
        im[i] = 0.0f;
    }
#if HAVE_ASYNC_LDS
    asm volatile("s_wait_asynccnt 0" ::: "memory");
#endif
    __syncthreads();

    // log-spectrum = FFT(ccep)
    fft1024_lds(re, im, tid);

    // spectrum = exp(log_spectrum); conjugate for inverse transform
    for (int i = tid; i < FFTN; i += 256) {
        const float er = __expf(re[i]);
        float si, ci;
        __sincosf(im[i], &si, &ci);
        re[i] =  er * ci;
        im[i] = -er * si;       // conj
    }
    __syncthreads();

    // h = real(ifft(spec)) = real(fft(conj(spec)))/N
    fft1024_lds(re, im, tid);
    for (int i = tid; i < FFTN; i += 256) re[i] *= (1.0f / (float)FFTN);
    __syncthreads();

    // Direct convolution, 5 output taps per thread (j = tid + 256*s).
    // j=tid and j=tid+1024 share one LDS read via (tid-k) mod 1024.
    float a0 = 0.f, a1 = 0.f, a2 = 0.f, a3 = 0.f, a4 = 0.f;
    for (int k = 0; k < HOPSZ; ++k) {
        const float f  = fr[k];
        const float v0 = re[(tid - k) & (FFTN - 1)];
        a1 = fmaf(f, re[tid + 256 - k], a1);
        a2 = fmaf(f, re[tid + 512 - k], a2);
        a3 = fmaf(f, re[tid + 768 - k], a3);
        const float fv = f * v0;
        if (k <= tid) a0 += fv; else a4 += fv;
    }

    const size_t base = (size_t)b * NSAMP;
    const int    g    = t * HOPSZ + tid;
    if (g               < NSAMP) atomicAdd(&out[base + g],        a0);
    if (g + 256         < NSAMP) atomicAdd(&out[base + g + 256],  a1);
    if (g + 512         < NSAMP) atomicAdd(&out[base + g + 512],  a2);
    if (g + 768         < NSAMP) atomicAdd(&out[base + g + 768],  a3);
    if (tid < 239 && g + 1024 < NSAMP) atomicAdd(&out[base + g + 1024], a4);
}

// ---------------------------------------------------------------------------
// Kernel 5: final clip to [-1, 1]
// ---------------------------------------------------------------------------
__global__ __launch_bounds__(256)
void clip_out(float* __restrict__ out, int n)
{
    const int i = blockIdx.x * 256 + threadIdx.x;
    if (i < n) out[i] = fminf(fmaxf(out[i], -1.0f), 1.0f);
}

// ---------------------------------------------------------------------------
extern "C" void kernel_launch(void* const* d_in, const int* in_sizes, int n_in,
                              void* d_out, int out_size, void* d_ws, size_t ws_size,
                              hipStream_t stream)
{
    const float* x   = (const float*)d_in[0];   // (16,1000,80)
    const float* cf0 = (const float*)d_in[1];   // (16,1,1000)
    const float* z   = (const float*)d_in[2];   // (16,1,240000)
    const float* Wh  = (const float*)d_in[3];   // (80,222)
    const float* bh  = (const float*)d_in[4];   // (222,)
    const float* Wn  = (const float*)d_in[5];   // (80,222)
    const float* bn  = (const float*)d_in[6];   // (222,)
    float* out = (float*)d_out;                 // (16,240000)

    char*  ws  = (char*)d_ws;
    size_t off = 0;
    auto alloc = [&](size_t bytes) -> void* {
        void* p = ws + off;
        off += (bytes + 255) & ~(size_t)255;
        return p;
    };
    float* ccepH = (float*)alloc((size_t)BATCH * TFRM * CPAD * sizeof(float));
    float* ccepN = (float*)alloc((size_t)BATCH * TFRM * CPAD * sizeof(float));
    float* pre   = (float*)alloc((size_t)BATCH * TFRM * sizeof(float));
    float* hsrc  = (float*)alloc((size_t)BATCH * NSAMP * sizeof(float));
    (void)ws_size; (void)in_sizes; (void)n_in;

    (void)hipMemsetAsync(out, 0, (size_t)out_size * sizeof(float), stream);

    // 125 m-groups x 14 n-tiles x 2 branches = 3500 blocks, 8 waves each
    gemm_ccep_wmma<<<MGROUPS * NTILES * 2, 256, 0, stream>>>(
        x, Wh, bh, Wn, bn, ccepH, ccepN);

    f0_prefix<<<1, 32, 0, stream>>>(cf0, pre);

    impulse_train<<<(BATCH * NSAMP + 255) / 256, 256, 0, stream>>>(cf0, pre, hsrc);

    ltv_filter<<<BATCH * TFRM * 2, 256, 0, stream>>>(ccepH, ccepN, hsrc, z, out);

    clip_out<<<(out_size + 255) / 256, 256, 0, stream>>>(out, out_size);
}